// Memory_67078799228945
// MI455X (gfx1250) — compile-verified
//
#include <hip/hip_runtime.h>
#include <hip/hip_bf16.h>

typedef __attribute__((ext_vector_type(2))) float v2f;
typedef __attribute__((ext_vector_type(4))) float v4f;
typedef __attribute__((ext_vector_type(8))) float v8f;

#define BATCH       65536
#define ROWS_PER_WG 64
#define THREADS     128

// Cubic B-spline basis, uniform grid g[t] = -1 + 0.4*(t-3), t=0..11, K=3 -> 8 values.
__device__ __forceinline__ void bspline8(float x, float out[8]) {
    const float h = 0.4f;
    float g[12];
#pragma unroll
    for (int t = 0; t < 12; ++t) g[t] = -1.0f + h * (float)(t - 3);

    float v0[11];
#pragma unroll
    for (int t = 0; t < 11; ++t)
        v0[t] = (x >= g[t] && x < g[t + 1]) ? 1.0f : 0.0f;

    const float r1 = 1.0f / h;
    float v1[10];
#pragma unroll
    for (int t = 0; t < 10; ++t)
        v1[t] = (x - g[t]) * r1 * v0[t] + (g[t + 2] - x) * r1 * v0[t + 1];

    const float r2 = 1.0f / (2.0f * h);
    float v2[9];
#pragma unroll
    for (int t = 0; t < 9; ++t)
        v2[t] = (x - g[t]) * r2 * v1[t] + (g[t + 3] - x) * r2 * v1[t + 1];

    const float r3 = 1.0f / (3.0f * h);
#pragma unroll
    for (int t = 0; t < 8; ++t)
        out[t] = (x - g[t]) * r3 * v2[t] + (g[t + 4] - x) * r3 * v2[t + 1];
}

// Pack (coef*ss, sb) into the B-operand layout for v_wmma_f32_16x16x4_f32 with
// lane-contiguous tile fragments:
//   Wp[((kp*16 + rl)*NT + tt)*2 + lo] = W[2*kp+lo][tt*16 + rl]
// where feature k = i*9+m (m<8: coef[i][n][m]*ss[i][n]; m==8: sb[i][n]),
// zero beyond din*9. A lane (rl) reads its NT tile pairs as contiguous 16B/32B.
__global__ void kan_pack_weights(const float* __restrict__ coef,
                                 const float* __restrict__ sb,
                                 const float* __restrict__ ss,
                                 float* __restrict__ Wp,
                                 int din, int dout, int kpairs) {
    int idx = blockIdx.x * blockDim.x + threadIdx.x;
    int total = kpairs * dout;
    if (idx >= total) return;
    int kp = idx / dout;
    int n  = idx - kp * dout;      // n = tt*16 + rl
    int rl = n & 15;
    int tt = n >> 4;
    int nt = dout >> 4;
    int kmax = din * 9;
#pragma unroll
    for (int lo = 0; lo < 2; ++lo) {
        int k = 2 * kp + lo;
        float v = 0.0f;
        if (k < kmax) {
            int i = k / 9;
            int m = k - i * 9;
            float s = ss[i * dout + n];
            v = (m < 8) ? coef[(i * dout + n) * 8 + m] * s
                        : sb[i * dout + n];
        }
        Wp[((kp * 16 + rl) * nt + tt) * 2 + lo] = v;
    }
}

// One KAN bank: features (spline basis + passthrough) -> LDS, then fp32 WMMA GEMM.
// Fully shape-templated: static tile loop, pinned accumulators, immediate-offset stores.
template <int DIN, int KP4, int DOUT, int LSTR, int OSTR>
__global__ __launch_bounds__(THREADS)
void kan_bank_kernel(const float* __restrict__ xin,
                     const float* __restrict__ hid,   // d_out hidden base (B x 512)
                     const float* __restrict__ Wp,
                     float* __restrict__ outp,
                     int bank, int colOff) {
    constexpr int NTILES  = DOUT / 16;
    constexpr int NCHUNKS = KP4 / 4;

    extern __shared__ float lds[];
    const int tid  = threadIdx.x;
    const int row0 = blockIdx.x * ROWS_PER_WG;

    // ---- Phase 1: build feature rows in LDS: f[r][i*9 + m] ----
    constexpr int TOTAL = ROWS_PER_WG * DIN;
    for (int idx = tid; idx < TOTAL; idx += THREADS) {
        int r = idx / DIN;
        int i = idx - r * DIN;
        int b = row0 + r;
        float hv;
        if (bank == 8)        hv = hid[b * 512 + 448 + i];
        else if (i == 0)      hv = xin[b * 8 + bank];
        else                  hv = hid[b * 512 + (bank - 1) * 64 + (i - 1)];
        float bs[8];
        bspline8(hv, bs);
        float* dst = &lds[r * LSTR + i * 9];
#pragma unroll
        for (int m = 0; m < 8; ++m) dst[m] = bs[m];
        dst[8] = hv;
    }
    // zero-pad tail features (k in [DIN*9, KP4))
    constexpr int PADBEG = DIN * 9;
    constexpr int PADW   = KP4 - PADBEG;
    if constexpr (PADW > 0) {
        for (int idx = tid; idx < ROWS_PER_WG * PADW; idx += THREADS) {
            int r = idx / PADW;
            int k = PADBEG + (idx - r * PADW);
            lds[r * LSTR + k] = 0.0f;
        }
    }
    __syncthreads();

    // ---- Phase 2: WMMA GEMM ----
    const int wave = tid >> 5;
    const int lane = tid & 31;
    const int rl   = lane & 15;   // row-in-tile (A) / col-in-tile (B,C)
    const int hi   = lane >> 4;   // K-half select

    v8f acc[NTILES];
#pragma unroll
    for (int t = 0; t < NTILES; ++t) acc[t] = v8f{};

    const float* arow = &lds[(wave * 16 + rl) * LSTR + 2 * hi];
    // lane's B base: K-pair row (2c+hi), lane column rl, NTILES contiguous pairs
    const float* bbas = Wp + (size_t)(hi * 16 + rl) * (2 * NTILES);

#pragma unroll 2
    for (int c = 0; c < NCHUNKS; ++c) {
        v2f a = *(const v2f*)(arow + 4 * c);
        const float* bp = bbas + (size_t)(2 * c) * (16 * 2 * NTILES);
        v2f bf[NTILES];
        if constexpr (NTILES == 4) {
            v4f b01 = *(const v4f*)(bp);
            v4f b23 = *(const v4f*)(bp + 4);
            bf[0] = v2f{b01.x, b01.y};
            bf[1] = v2f{b01.z, b01.w};
            bf[2] = v2f{b23.x, b23.y};
            bf[3] = v2f{b23.z, b23.w};
        } else {
            bf[0] = *(const v2f*)(bp);
        }
#pragma unroll
        for (int t = 0; t < NTILES; ++t)
            acc[t] = __builtin_amdgcn_wmma_f32_16x16x4_f32(
                false, a, false, bf[t], (short)0, acc[t], false, false);
    }

    // ---- Store C tiles: lane (rl,hi) owns col rl of rows hi*8+v ----
    float* obase = outp + (size_t)(row0 + wave * 16 + hi * 8) * OSTR + colOff + rl;
#pragma unroll
    for (int t = 0; t < NTILES; ++t) {
#pragma unroll
        for (int v = 0; v < 8; ++v)
            obase[v * OSTR + 16 * t] = acc[t][v];
    }
}

extern "C" void kernel_launch(void* const* d_in, const int* in_sizes, int n_in,
                              void* d_out, int out_size, void* d_ws, size_t ws_size,
                              hipStream_t stream) {
    const float* xin = (const float*)d_in[0];
    const float* coefs[9];
    const float* sbs[9];
    const float* sss[9];
    for (int i = 0; i < 9; ++i) {
        coefs[i] = (const float*)d_in[1 + i];
        sbs[i]   = (const float*)d_in[10 + i];
        sss[i]   = (const float*)d_in[19 + i];
    }
    float* out = (float*)d_out;
    float* ws  = (float*)d_ws;

    static const int DIN[9]  = {1, 65, 65, 65, 65, 65, 65, 65, 64};
    static const int DOUT[9] = {64, 64, 64, 64, 64, 64, 64, 64, 16};
    static const int KP4[9]  = {12, 588, 588, 588, 588, 588, 588, 588, 576};
    static const int LSTR[9] = {12, 588, 588, 588, 588, 588, 588, 588, 580};

    size_t wpoff[9];
    size_t off = 0;
    for (int b = 0; b < 9; ++b) {
        int kpairs = KP4[b] / 2;
        int total  = kpairs * DOUT[b];
        wpoff[b] = off;
        kan_pack_weights<<<(total + 255) / 256, 256, 0, stream>>>(
            coefs[b], sbs[b], sss[b], ws + off, DIN[b], DOUT[b], kpairs);
        off += (size_t)total * 2;
    }

    const int nblocks = BATCH / ROWS_PER_WG;
    for (int b = 0; b < 9; ++b) {
        float* outp;
        int coloff;
        if (b < 8) { outp = out;                       coloff = b * 64; }
        else       { outp = out + (size_t)BATCH * 512; coloff = 0; }
        size_t ldsBytes = (size_t)ROWS_PER_WG * LSTR[b] * sizeof(float);
        if (b == 0) {
            kan_bank_kernel<1, 12, 64, 12, 512><<<nblocks, THREADS, ldsBytes, stream>>>(
                xin, out, ws + wpoff[b], outp, b, coloff);
        } else if (b < 8) {
            kan_bank_kernel<65, 588, 64, 588, 512><<<nblocks, THREADS, ldsBytes, stream>>>(
                xin, out, ws + wpoff[b], outp, b, coloff);
        } else {
            kan_bank_kernel<64, 576, 16, 580, 16><<<nblocks, THREADS, ldsBytes, stream>>>(
                xin, out, ws + wpoff[b], outp, b, coloff);
        }
    }
}